// PtModule_76166950027566
// MI455X (gfx1250) — compile-verified
//
#include <hip/hip_runtime.h>

// y[i] = x[i] * 2 + 5  -- streaming affine map, HBM-bandwidth bound.
// 512 MiB of traffic @ 23.3 TB/s => ~23 us floor; compute is negligible.
// Path the MI455X rewards: b128 non-temporal loads/stores (footprint >> 192MB
// L2, zero reuse), 32-bit GVS addressing with immediate offsets, branch-free
// fast path for full blocks, wave32-friendly 256-thread blocks.

typedef float v4f __attribute__((ext_vector_type(4)));

#define VECS_PER_THREAD 4
#define BLOCK_THREADS   256

__global__ __launch_bounds__(BLOCK_THREADS)
void affine2p5_kernel(const v4f* __restrict__ xv, v4f* __restrict__ yv,
                      int n4, const float* __restrict__ x,
                      float* __restrict__ y, int n) {
    int base = blockIdx.x * (BLOCK_THREADS * VECS_PER_THREAD) + (int)threadIdx.x;

    if (base + (VECS_PER_THREAD - 1) * BLOCK_THREADS < n4) {
        // Fast path: fully in-bounds. Unguarded b128 NT loads with immediate
        // offsets (k * 256 float4 = k * 4096 bytes), then FMA, then NT stores.
        v4f v[VECS_PER_THREAD];
#pragma unroll
        for (int k = 0; k < VECS_PER_THREAD; ++k)
            v[k] = __builtin_nontemporal_load(xv + base + k * BLOCK_THREADS);
#pragma unroll
        for (int k = 0; k < VECS_PER_THREAD; ++k) {
            v4f r = v[k] * 2.0f + 5.0f;   // contracts to per-lane FMA
            __builtin_nontemporal_store(r, yv + base + k * BLOCK_THREADS);
        }
    } else {
        // Boundary block only: guarded per-vector.
#pragma unroll
        for (int k = 0; k < VECS_PER_THREAD; ++k) {
            int i = base + k * BLOCK_THREADS;
            if (i < n4) {
                v4f r = __builtin_nontemporal_load(xv + i) * 2.0f + 5.0f;
                __builtin_nontemporal_store(r, yv + i);
            }
        }
        // Scalar tail for n % 4 (not hit for 8192*8192, kept for correctness).
        if (blockIdx.x == 0) {
            int tail = n - n4 * 4;
            if ((int)threadIdx.x < tail) {
                int i = n4 * 4 + (int)threadIdx.x;
                y[i] = __builtin_fmaf(x[i], 2.0f, 5.0f);
            }
        }
    }
}

extern "C" void kernel_launch(void* const* d_in, const int* in_sizes, int n_in,
                              void* d_out, int out_size, void* d_ws, size_t ws_size,
                              hipStream_t stream) {
    const float* x = (const float*)d_in[0];
    float* y = (float*)d_out;

    int n  = in_sizes[0];      // 8192*8192 = 67,108,864
    int n4 = n >> 2;           // float4 count (16,777,216)

    int vecs_per_block = BLOCK_THREADS * VECS_PER_THREAD;
    int blocks = (n4 + vecs_per_block - 1) / vecs_per_block;
    if (blocks < 1) blocks = 1;

    affine2p5_kernel<<<blocks, BLOCK_THREADS, 0, stream>>>(
        (const v4f*)x, (v4f*)y, n4, x, y, n);
}